// Router_18777597018867
// MI455X (gfx1250) — compile-verified
//
#include <hip/hip_runtime.h>
#include <hip/hip_bf16.h>
#include <math.h>

typedef __attribute__((ext_vector_type(2))) float v2f;
typedef __attribute__((ext_vector_type(8))) float v8f;

#define N_TOK   32768
#define D_DIM   1024
#define E_EXP   16
#define ROWPAD  1028          // 1024 + 4-float pad -> LDS bank stride of 4
#define WAVES_PER_WG 8
#define THREADS 256
#define TILES_PER_WAVE 2      // two 16-token tiles per wave share B fragments

__global__ void zero_load_kernel(float* __restrict__ load_out) {
    if (threadIdx.x < E_EXP) load_out[threadIdx.x] = 0.0f;
}

// Per-16-token-tile epilogue: softplus/noise mix, softmax over 16 experts
// (16-lane groups), top-2 butterfly, writes gates/indices, accumulates load.
__device__ __forceinline__ void epilogue(const v8f& accG, const v8f& accN,
                                         int tileBase, int h, int e,
                                         const float* __restrict__ noise,
                                         float* __restrict__ gates_out,
                                         int*   __restrict__ idx_out,
                                         float& loadAcc) {
    #pragma unroll
    for (int r = 0; r < 8; ++r) {
        const int tok = tileBase + r + 8 * h;
        float clean = accG[r];
        float nl    = accN[r];
        float sp    = (nl > 20.0f) ? nl : log1pf(expf(nl));       // softplus
        float nz    = noise[(size_t)tok * E_EXP + e];
        float logit = clean + nz * sp * 0.01f;

        // softmax across 16 lanes of this half-wave (masks <16 stay in-group)
        float m = logit;
        m = fmaxf(m, __shfl_xor(m, 1));
        m = fmaxf(m, __shfl_xor(m, 2));
        m = fmaxf(m, __shfl_xor(m, 4));
        m = fmaxf(m, __shfl_xor(m, 8));
        float p = expf(logit - m);
        float s = p;
        s += __shfl_xor(s, 1);
        s += __shfl_xor(s, 2);
        s += __shfl_xor(s, 4);
        s += __shfl_xor(s, 8);
        float gate = p / s;
        loadAcc += gate;

        // top-2 butterfly merge across the 16-lane group
        float v1 = gate, v2 = -1.0f;   // gates >= 0 so -1 acts as -inf
        int   i1 = e,    i2 = 0;
        #pragma unroll
        for (int msk = 1; msk <= 8; msk <<= 1) {
            float b1v = __shfl_xor(v1, msk);
            int   b1i = __shfl_xor(i1, msk);
            float b2v = __shfl_xor(v2, msk);
            int   b2i = __shfl_xor(i2, msk);
            if (b1v > v1) {
                float na2; int ni2;
                if (v1 > b2v) { na2 = v1;  ni2 = i1;  }
                else          { na2 = b2v; ni2 = b2i; }
                v1 = b1v; i1 = b1i; v2 = na2; i2 = ni2;
            } else if (b1v > v2) {
                v2 = b1v; i2 = b1i;
            }
        }

        if (e == 0) {
            float denom = v1 + v2 + 1e-8f;
            gates_out[(size_t)tok * 2 + 0] = v1 / denom;
            gates_out[(size_t)tok * 2 + 1] = v2 / denom;
            idx_out[(size_t)tok * 2 + 0] = i1;
            idx_out[(size_t)tok * 2 + 1] = i2;
        }
    }
}

__global__ __launch_bounds__(THREADS)
void router_kernel(const float* __restrict__ x,
                   const float* __restrict__ noise,
                   const float* __restrict__ Wg,
                   const float* __restrict__ Wn,
                   float* __restrict__ gates_out,
                   int*   __restrict__ idx_out,
                   float* __restrict__ load_out) {
    __shared__ float ldsG[E_EXP * ROWPAD];
    __shared__ float ldsN[E_EXP * ROWPAD];

    // Stage W_gate / W_noise into LDS (padded rows -> conflict-free B reads).
    for (int i = threadIdx.x; i < E_EXP * (D_DIM / 4); i += THREADS) {
        int row = i >> 8;          // 256 float4 per 1024-float row
        int c4  = i & 255;
        float4 g = ((const float4*)(Wg + (size_t)row * D_DIM))[c4];
        float4 n = ((const float4*)(Wn + (size_t)row * D_DIM))[c4];
        *(float4*)&ldsG[row * ROWPAD + c4 * 4] = g;
        *(float4*)&ldsN[row * ROWPAD + c4 * 4] = n;
    }
    __syncthreads();

    const int lane = threadIdx.x & 31;
    const int wave = threadIdx.x >> 5;
    const int h    = lane >> 4;      // half-wave group (K slot group)
    const int l15  = lane & 15;

    // Each wave owns TWO 16-token tiles; B fragments are shared between them.
    const int pair  = blockIdx.x * WAVES_PER_WG + wave;       // 1024 pairs
    const int tok0  = pair * (16 * TILES_PER_WAVE) + l15;     // A row, tile 0
    const int tok1  = tok0 + 16;                              // A row, tile 1

    const float* xrow0 = x + (size_t)tok0 * D_DIM;
    const float* xrow1 = x + (size_t)tok1 * D_DIM;
    const float* bgrow = &ldsG[l15 * ROWPAD];
    const float* bnrow = &ldsN[l15 * ROWPAD];

    v8f accG0 = {0.f,0.f,0.f,0.f,0.f,0.f,0.f,0.f};
    v8f accN0 = {0.f,0.f,0.f,0.f,0.f,0.f,0.f,0.f};
    v8f accG1 = {0.f,0.f,0.f,0.f,0.f,0.f,0.f,0.f};
    v8f accN1 = {0.f,0.f,0.f,0.f,0.f,0.f,0.f,0.f};

    // K loop: per iteration each lane loads one float4 of x per tile and one
    // float4 of each W row; that covers two 16x16x4 WMMA K-steps (K=8/iter).
    // A and B use the same (half, slot)->physical-K mapping, so the reduction
    // over K is exact regardless of the hardware's internal K ordering.
    #pragma unroll 2
    for (int k = 0; k < D_DIM; k += 8) {
        float4 xa = *(const float4*)(xrow0 + k + 4 * h);
        float4 xb = *(const float4*)(xrow1 + k + 4 * h);
        float4 bg = *(const float4*)(bgrow + k + 4 * h);
        float4 bn = *(const float4*)(bnrow + k + 4 * h);
        v2f a0 = {xa.x, xa.y}, a1 = {xa.z, xa.w};
        v2f c0 = {xb.x, xb.y}, c1 = {xb.z, xb.w};
        v2f g0 = {bg.x, bg.y}, g1 = {bg.z, bg.w};
        v2f n0 = {bn.x, bn.y}, n1 = {bn.z, bn.w};
        accG0 = __builtin_amdgcn_wmma_f32_16x16x4_f32(false, a0, false, g0, (short)0, accG0, false, false);
        accN0 = __builtin_amdgcn_wmma_f32_16x16x4_f32(false, a0, false, n0, (short)0, accN0, false, false);
        accG1 = __builtin_amdgcn_wmma_f32_16x16x4_f32(false, c0, false, g0, (short)0, accG1, false, false);
        accN1 = __builtin_amdgcn_wmma_f32_16x16x4_f32(false, c0, false, n0, (short)0, accN1, false, false);
        accG0 = __builtin_amdgcn_wmma_f32_16x16x4_f32(false, a1, false, g1, (short)0, accG0, false, false);
        accN0 = __builtin_amdgcn_wmma_f32_16x16x4_f32(false, a1, false, n1, (short)0, accN0, false, false);
        accG1 = __builtin_amdgcn_wmma_f32_16x16x4_f32(false, c1, false, g1, (short)0, accG1, false, false);
        accN1 = __builtin_amdgcn_wmma_f32_16x16x4_f32(false, c1, false, n1, (short)0, accN1, false, false);
    }

    const int e = l15;
    float loadAcc = 0.0f;
    epilogue(accG0, accN0, pair * (16 * TILES_PER_WAVE),      h, e,
             noise, gates_out, idx_out, loadAcc);
    epilogue(accG1, accN1, pair * (16 * TILES_PER_WAVE) + 16, h, e,
             noise, gates_out, idx_out, loadAcc);

    // load[e]: combine half-wave partials, one atomic per expert per wave
    loadAcc += __shfl_xor(loadAcc, 16);
    if (lane < 16) atomicAdd(&load_out[e], loadAcc);
}

extern "C" void kernel_launch(void* const* d_in, const int* in_sizes, int n_in,
                              void* d_out, int out_size, void* d_ws, size_t ws_size,
                              hipStream_t stream) {
    const float* x     = (const float*)d_in[0];   // (8,4096,1024)
    const float* noise = (const float*)d_in[1];   // (32768,16)
    const float* Wg    = (const float*)d_in[2];   // (16,1024)
    const float* Wn    = (const float*)d_in[3];   // (16,1024)

    float* out       = (float*)d_out;
    float* gates_out = out;                                  // N*2 floats
    int*   idx_out   = (int*)(out + (size_t)N_TOK * 2);      // N*2 ints (bit pattern)
    float* load_out  = out + (size_t)N_TOK * 4;              // 16 floats

    zero_load_kernel<<<1, 32, 0, stream>>>(load_out);

    const int pairs = N_TOK / (16 * TILES_PER_WAVE);         // 1024 wave-jobs
    const int grid  = pairs / WAVES_PER_WG;                  // 128 workgroups
    router_kernel<<<grid, THREADS, 0, stream>>>(x, noise, Wg, Wn,
                                                gates_out, idx_out, load_out);
}